// BatchHide_88295937671575
// MI455X (gfx1250) — compile-verified
//
#include <hip/hip_runtime.h>
#include <cstdint>
#include <cstddef>

namespace {

constexpr int kB = 32, kC = 128, kH = 224, kW = 224;
constexpr int kHW  = kH * kW;               // 50176
constexpr int kHW4 = kHW / 4;               // 12544 float4 per image plane
constexpr int kTPB = 256;                   // 8 wave32 per block
constexpr int kBlocksPerImg = kHW4 / kTPB;  // 49 (exact -> no bounds checks)
constexpr int kGrid = kB * kBlocksPerImg;   // 1568

typedef float v4f __attribute__((ext_vector_type(4)));

__global__ __launch_bounds__(kTPB) void gridmask_mul_kernel(
    const float* __restrict__ feature,
    const float* __restrict__ mask,
    float* __restrict__ out)
{
  __shared__ v4f smem[kTPB];   // 4 KB mask tile for this block

  const int b   = blockIdx.x / kBlocksPerImg;   // image index (uniform per block)
  const int blk = blockIdx.x % kBlocksPerImg;   // tile index within image
  const int hw4 = blk * kTPB + threadIdx.x;     // float4 index within H*W

  // ---- Stage this block's 4 KB mask tile through LDS via the CDNA5
  //      async DMA path (GLOBAL_LOAD_ASYNC_TO_LDS_B128, tracked by ASYNCcnt).
  //      The tile is reused by all 128 channel iterations below.
  {
    const float* gmask = mask + (size_t)b * kHW + (size_t)blk * kTPB * 4; // uniform -> SGPR pair
    unsigned gvoff = threadIdx.x * 16u;                                   // byte offset in tile
    unsigned ldsa  = (unsigned)(uintptr_t)(&smem[0]) + threadIdx.x * 16u; // LDS byte address
    asm volatile("global_load_async_to_lds_b128 %0, %1, %2"
                 :: "v"(ldsa), "v"(gvoff), "s"(gmask) : "memory");
    asm volatile("s_wait_asynccnt 0" ::: "memory");
  }
  const v4f m = smem[threadIdx.x];   // mask float4, register-resident for the whole loop

  // Base offset in v4f units; channel stride = kHW4 v4f = 200704 B,
  // so unroll-by-8 offsets (<= 1.4 MB) fit the signed 24-bit IOFFSET.
  const size_t base = ((size_t)b * kC) * kHW4 + (size_t)hw4;
  const v4f* __restrict__ f = reinterpret_cast<const v4f*>(feature) + base;
  v4f*       __restrict__ o = reinterpret_cast<v4f*>(out) + base;

#pragma unroll 8
  for (int c = 0; c < kC; ++c) {
    v4f x = __builtin_nontemporal_load(f);   // streamed once: NT load
    __builtin_nontemporal_store(x * m, o);   // streamed once: NT store
    f += kHW4;
    o += kHW4;
  }
}

} // namespace

extern "C" void kernel_launch(void* const* d_in, const int* in_sizes, int n_in,
                              void* d_out, int out_size, void* d_ws, size_t ws_size,
                              hipStream_t stream) {
  (void)in_sizes; (void)n_in; (void)out_size; (void)d_ws; (void)ws_size;
  const float* feature = (const float*)d_in[0];   // [B,C,H,W] fp32
  const float* mask    = (const float*)d_in[1];   // [B,1,H,W] fp32
  float* out = (float*)d_out;                     // [B,C,H,W] fp32
  gridmask_mul_kernel<<<kGrid, kTPB, 0, stream>>>(feature, mask, out);
}